// DensePoseV1ConvXGNSparseGNHead_25683904430833
// MI455X (gfx1250) — compile-verified
//
#include <hip/hip_runtime.h>
#include <math.h>

typedef __attribute__((ext_vector_type(2))) float v2f;
typedef __attribute__((ext_vector_type(8))) float v8f;

#define NCH   64
#define NINS  64
#define EPSN  1e-5f

// ---- workspace layout (float slots) ----
// 0     : s_g   [4096]   atomic-accumulated sum(x)
// 4096  : s2_g  [4096]   atomic-accumulated sum(x*x)
// 8192  : cnt_g [64]     (uint) per-instance counts
// 8256  : mu_g  [4096]
// 12352 : inv_g [4096]   rsqrt(var+eps)
// 16448 : m_g   [4096]   atomic-accumulated sum(relu(xn))
// 20544 : w_g   [4096]   sigmoid(conv) per (instance,channel)
#define WS_FLOATS 24640

// ---------------- per-instance point counts ----------------
__global__ void k_count(const int* __restrict__ seg, int n, unsigned* __restrict__ cnt_g) {
    __shared__ unsigned c[NINS];
    int t = threadIdx.x;
    if (t < NINS) c[t] = 0u;
    __syncthreads();
    for (int i = blockIdx.x * blockDim.x + t; i < n; i += gridDim.x * blockDim.x)
        atomicAdd(&c[seg[i]], 1u);
    __syncthreads();
    if (t < NINS) atomicAdd(&cnt_g[t], c[t]);
}

// ---------------- pass 1: segment sums of x and x^2 via one-hot WMMA ----------------
// Block = 256 threads = 8 waves. wave: ct = channel tile (0..3), pw = point-wave (0..1).
// Straight-line inner loop, unconditional clamped loads (OOB rows get one-hot 0,
// value is finite so it contributes exactly 0 to the WMMA accumulate).
__global__ __launch_bounds__(256) void k_stats(const float* __restrict__ f,
                                               const int* __restrict__ seg, int n,
                                               float* __restrict__ s_g,
                                               float* __restrict__ s2_g) {
    __shared__ float red[2 * 4096];
    int tid = threadIdx.x;
    for (int i = tid; i < 8192; i += 256) red[i] = 0.f;
    __syncthreads();

    int wave = tid >> 5;
    int lane = tid & 31;
    int ct   = wave & 3;      // channel tile
    int pw   = wave >> 2;     // point-wave within block (0..1)
    int q    = lane & 15;
    int hi   = lane >> 4;     // half-wave id
    int col  = ct * 16 + q;   // global channel
    int nm1  = n - 1;

    v8f accs[4], acc2[4];
#pragma unroll
    for (int t = 0; t < 4; ++t) { accs[t] = v8f{}; acc2[t] = v8f{}; }

    int ngroups = (n + 3) >> 2;
    int gstride = gridDim.x * 2;
    for (int g = blockIdx.x * 2 + pw; g < ngroups; g += gstride) {
        int rowA = g * 4 + hi * 2;            // lanes 0-15: K0 point, lanes 16-31: K2
        int rowB = rowA + 1;                  // K1 / K3 point
        int rA = rowA < nm1 ? rowA : nm1;     // clamped, always in-bounds
        int rB = rowB < nm1 ? rowB : nm1;
        int sA = seg[rA];
        int sB = seg[rB];
        sA = (rowA < n) ? sA : -1;            // OOB -> one-hot column all zero
        sB = (rowB < n) ? sB : -1;
        v2f b;
        b.x = f[(size_t)rA * NCH + col];      // finite even when clamped
        b.y = f[(size_t)rB * NCH + col];
        v2f b2; b2.x = b.x * b.x; b2.y = b.y * b.y;

        // prefetch next group's rows into GL2 (speculative; safe if OOB)
        __builtin_prefetch(f + (size_t)(rowA + gstride * 4) * NCH + col, 0, 0);

#pragma unroll
        for (int t = 0; t < 4; ++t) {
            int base = t * 16;
            v2f a;
            a.x = (sA == base + q) ? 1.f : 0.f;
            a.y = (sB == base + q) ? 1.f : 0.f;
            accs[t] = __builtin_amdgcn_wmma_f32_16x16x4_f32(
                false, a, false, b,  (short)0, accs[t], false, false);
            acc2[t] = __builtin_amdgcn_wmma_f32_16x16x4_f32(
                false, a, false, b2, (short)0, acc2[t], false, false);
        }
    }

    // merge register tiles into LDS, then LDS -> tiny global tables
#pragma unroll
    for (int t = 0; t < 4; ++t) {
#pragma unroll
        for (int r = 0; r < 8; ++r) {
            int m = t * 16 + r + hi * 8;      // C/D layout: VGPR r = row r (+8 for hi)
            atomicAdd(&red[m * NCH + col],        accs[t][r]);
            atomicAdd(&red[4096 + m * NCH + col], acc2[t][r]);
        }
    }
    __syncthreads();
    for (int i = tid; i < 4096; i += 256) {
        atomicAdd(&s_g[i],  red[i]);
        atomicAdd(&s2_g[i], red[4096 + i]);
    }
}

// ---------------- finalize mu / inv-std ----------------
__global__ void k_finalize(const float* __restrict__ s_g, const float* __restrict__ s2_g,
                           const unsigned* __restrict__ cnt_g,
                           float* __restrict__ mu_g, float* __restrict__ inv_g) {
    int i = blockIdx.x * blockDim.x + threadIdx.x;
    if (i >= 4096) return;
    float c   = fmaxf((float)cnt_g[i >> 6], 1.f);
    float mu  = s_g[i] / c;
    float var = s2_g[i] / c - mu * mu;
    mu_g[i]  = mu;
    inv_g[i] = rsqrtf(var + EPSN);
}

// ---------------- pass 2: segment sums of relu(xn) via one-hot WMMA ----------------
__global__ __launch_bounds__(256) void k_xrsum(const float* __restrict__ f,
                                               const int* __restrict__ seg, int n,
                                               const float* __restrict__ mu_g,
                                               const float* __restrict__ inv_g,
                                               float* __restrict__ m_g) {
    __shared__ float red[4096];
    __shared__ float mu_s[4096];
    __shared__ float in_s[4096];
    int tid = threadIdx.x;
    for (int i = tid; i < 4096; i += 256) {
        red[i]  = 0.f;
        mu_s[i] = mu_g[i];
        in_s[i] = inv_g[i];
    }
    __syncthreads();

    int wave = tid >> 5;
    int lane = tid & 31;
    int ct   = wave & 3;
    int pw   = wave >> 2;
    int q    = lane & 15;
    int hi   = lane >> 4;
    int col  = ct * 16 + q;
    int nm1  = n - 1;

    v8f accm[4];
#pragma unroll
    for (int t = 0; t < 4; ++t) accm[t] = v8f{};

    int ngroups = (n + 3) >> 2;
    int gstride = gridDim.x * 2;
    for (int g = blockIdx.x * 2 + pw; g < ngroups; g += gstride) {
        int rowA = g * 4 + hi * 2;
        int rowB = rowA + 1;
        int rA = rowA < nm1 ? rowA : nm1;
        int rB = rowB < nm1 ? rowB : nm1;
        int sA = seg[rA];                     // valid segment (0..63) even when clamped
        int sB = seg[rB];
        v2f b;
        b.x = f[(size_t)rA * NCH + col];
        b.y = f[(size_t)rB * NCH + col];

        __builtin_prefetch(f + (size_t)(rowA + gstride * 4) * NCH + col, 0, 0);

        int iA = sA * NCH + col;
        int iB = sB * NCH + col;
        v2f xr;                               // finite; OOB rows neutralized via one-hot
        xr.x = fmaxf(0.f, (b.x - mu_s[iA]) * in_s[iA]);
        xr.y = fmaxf(0.f, (b.y - mu_s[iB]) * in_s[iB]);

        sA = (rowA < n) ? sA : -1;
        sB = (rowB < n) ? sB : -1;

#pragma unroll
        for (int t = 0; t < 4; ++t) {
            int base = t * 16;
            v2f a;
            a.x = (sA == base + q) ? 1.f : 0.f;
            a.y = (sB == base + q) ? 1.f : 0.f;
            accm[t] = __builtin_amdgcn_wmma_f32_16x16x4_f32(
                false, a, false, xr, (short)0, accm[t], false, false);
        }
    }

#pragma unroll
    for (int t = 0; t < 4; ++t) {
#pragma unroll
        for (int r = 0; r < 8; ++r) {
            int m = t * 16 + r + hi * 8;
            atomicAdd(&red[m * NCH + col], accm[t][r]);
        }
    }
    __syncthreads();
    for (int i = tid; i < 4096; i += 256)
        atomicAdd(&m_g[i], red[i]);
}

// ---------------- ECA: channel mean -> conv1d(k=3, pad 1) -> sigmoid ----------------
__global__ void k_eca(const float* __restrict__ m_g, const unsigned* __restrict__ cnt_g,
                      const float* __restrict__ wconv, float* __restrict__ w_g) {
    __shared__ float m_s[4096];
    int tid = threadIdx.x;
    for (int i = tid; i < 4096; i += 256) {
        float c = fmaxf((float)cnt_g[i >> 6], 1.f);
        m_s[i] = m_g[i] / c;
    }
    __syncthreads();
    float w0 = wconv[0], w1 = wconv[1], w2 = wconv[2];
    for (int i = tid; i < 4096; i += 256) {
        int c = i & (NCH - 1);
        float left  = (c > 0)       ? m_s[i - 1] : 0.f;
        float right = (c < NCH - 1) ? m_s[i + 1] : 0.f;
        float conv  = w0 * left + w1 * m_s[i] + w2 * right;
        w_g[i] = 1.f / (1.f + __expf(-conv));
    }
}

// ---------------- pass 3: out = relu(xn) * w[seg] (recompute xn, stream float4) ------
__global__ __launch_bounds__(256) void k_out(const float* __restrict__ f,
                                             const int* __restrict__ seg, int n,
                                             const float* __restrict__ mu_g,
                                             const float* __restrict__ inv_g,
                                             const float* __restrict__ w_g,
                                             float* __restrict__ out) {
    __shared__ float mu_s[4096], in_s[4096], w_s[4096];
    int tid = threadIdx.x;
    for (int i = tid; i < 4096; i += 256) {
        mu_s[i] = mu_g[i]; in_s[i] = inv_g[i]; w_s[i] = w_g[i];
    }
    __syncthreads();

    int wave = tid >> 5;
    int lane = tid & 31;
    int hi   = lane >> 4;
    int q    = lane & 15;       // float4 index within the 64-channel row
    int pstride = gridDim.x * 16;
    for (int p = blockIdx.x * 16 + wave * 2 + hi; p < n; p += pstride) {
        int s = seg[p];
        const float4* fr  = (const float4*)(f   + (size_t)p * NCH);
        float4*       orw = (float4*)(out + (size_t)p * NCH);
        __builtin_prefetch(f + (size_t)(p + pstride) * NCH, 0, 0);
        float4 x = fr[q];
        int b = s * NCH + q * 4;
        float4 r;
        r.x = fmaxf(0.f, (x.x - mu_s[b + 0]) * in_s[b + 0]) * w_s[b + 0];
        r.y = fmaxf(0.f, (x.y - mu_s[b + 1]) * in_s[b + 1]) * w_s[b + 1];
        r.z = fmaxf(0.f, (x.z - mu_s[b + 2]) * in_s[b + 2]) * w_s[b + 2];
        r.w = fmaxf(0.f, (x.w - mu_s[b + 3]) * in_s[b + 3]) * w_s[b + 3];
        orw[q] = r;
    }
}

extern "C" void kernel_launch(void* const* d_in, const int* in_sizes, int n_in,
                              void* d_out, int out_size, void* d_ws, size_t ws_size,
                              hipStream_t stream) {
    const float* f   = (const float*)d_in[0];
    const int*   seg = (const int*)d_in[1];
    const float* eca = (const float*)d_in[2];
    float*       out = (float*)d_out;
    int n = in_sizes[0] / NCH;

    float*    ws    = (float*)d_ws;
    float*    s_g   = ws;
    float*    s2_g  = ws + 4096;
    unsigned* cnt_g = (unsigned*)(ws + 8192);
    float*    mu_g  = ws + 8256;
    float*    inv_g = ws + 12352;
    float*    m_g   = ws + 16448;
    float*    w_g   = ws + 20544;

    size_t zbytes = (size_t)WS_FLOATS * sizeof(float);
    if (zbytes > ws_size) zbytes = ws_size;
    hipMemsetAsync(d_ws, 0, zbytes, stream);

    k_count   <<<512, 256, 0, stream>>>(seg, n, cnt_g);
    k_stats   <<<512, 256, 0, stream>>>(f, seg, n, s_g, s2_g);
    k_finalize<<<16,  256, 0, stream>>>(s_g, s2_g, cnt_g, mu_g, inv_g);
    k_xrsum   <<<512, 256, 0, stream>>>(f, seg, n, mu_g, inv_g, m_g);
    k_eca     <<<1,   256, 0, stream>>>(m_g, cnt_g, eca, w_g);
    k_out     <<<1024,256, 0, stream>>>(f, seg, n, mu_g, inv_g, w_g, out);
}